// GCN_2_53884659695770
// MI455X (gfx1250) — compile-verified
//
#include <hip/hip_runtime.h>
#include <math.h>

typedef __attribute__((ext_vector_type(2))) float v2f;
typedef __attribute__((ext_vector_type(8))) float v8f;

#define N_NODES 100000
#define D_IN    500
#define D_H     64
#define D_OUT   47
#define N_LAYER 4
#define ALPHA_C 0.1f

// ---------------- degree / normalization ----------------------------------

__global__ void deg_init_kernel(float* __restrict__ deg) {
    int i = blockIdx.x * blockDim.x + threadIdx.x;
    if (i < N_NODES) deg[i] = 1.0f;             // self-loop contributes 1
}

__global__ void deg_count_kernel(float* __restrict__ deg,
                                 const int* __restrict__ col, int e) {
    int i = blockIdx.x * blockDim.x + threadIdx.x;
    if (i < e) atomicAdd(&deg[col[i]], 1.0f);
}

__global__ void dinv_kernel(float* __restrict__ deg) {
    int i = blockIdx.x * blockDim.x + threadIdx.x;
    if (i < N_NODES) deg[i] = rsqrtf(deg[i]);   // deg >= 1 always
}

// ---------------- input GEMM: x0 = x @ W_in^T + b_in  (WMMA f32) ----------
// Wave = 16-row strip x 64 cols. A: x[m][k] (row-major, ld=500).
// B[k][n] = W_in[n*500 + k].

__global__ void __launch_bounds__(256)
gemm_in_kernel(const float* __restrict__ x, const float* __restrict__ Win,
               const float* __restrict__ bin, float* __restrict__ x0) {
    const int lane  = threadIdx.x & 31;
    const int wave  = threadIdx.x >> 5;
    const int row0  = (blockIdx.x * 8 + wave) * 16;
    const int mlo   = lane & 15;
    const int khalf = (lane >> 4) * 2;          // 0 or 2

    int mA = row0 + mlo;
    if (mA >= N_NODES) mA = N_NODES - 1;        // branchless clamp; stores guarded
    const float* arow = x + (size_t)mA * D_IN;

    v8f acc[4];
    acc[0] = (v8f){}; acc[1] = (v8f){}; acc[2] = (v8f){}; acc[3] = (v8f){};

    for (int k = 0; k < D_IN; k += 4) {
        v2f a;
        a.x = arow[k + khalf];
        a.y = arow[k + khalf + 1];
#pragma unroll
        for (int t = 0; t < 4; ++t) {
            const float* w = Win + (size_t)(t * 16 + mlo) * D_IN + k + khalf;
            v2f b; b.x = w[0]; b.y = w[1];
            acc[t] = __builtin_amdgcn_wmma_f32_16x16x4_f32(
                false, a, false, b, (short)0, acc[t], false, false);
        }
    }

    const int mbase = row0 + (lane >> 4) * 8;   // C/D: lanes16-31 hold M=r+8
#pragma unroll
    for (int t = 0; t < 4; ++t) {
        const int n = t * 16 + mlo;
        const float bias = bin[n];
#pragma unroll
        for (int r = 0; r < 8; ++r) {
            int m = mbase + r;
            if (m < N_NODES) x0[(size_t)m * D_H + n] = acc[t][r] + bias;
        }
    }
}

// ---------------- layer GEMM: h = (0.9*agg + 0.1*x0) @ W  (+ReLU) ---------
// B[k][n] = W[k*64 + n] (row-major 64x64).

__global__ void __launch_bounds__(256)
gemm_layer_kernel(const float* __restrict__ agg, const float* __restrict__ x0,
                  const float* __restrict__ W, float* __restrict__ hout,
                  int relu) {
    const int lane  = threadIdx.x & 31;
    const int wave  = threadIdx.x >> 5;
    const int row0  = (blockIdx.x * 8 + wave) * 16;
    const int mlo   = lane & 15;
    const int khalf = (lane >> 4) * 2;

    int mA = row0 + mlo;
    if (mA >= N_NODES) mA = N_NODES - 1;
    const float* ar = agg + (size_t)mA * D_H;
    const float* xr = x0  + (size_t)mA * D_H;

    v8f acc[4];
    acc[0] = (v8f){}; acc[1] = (v8f){}; acc[2] = (v8f){}; acc[3] = (v8f){};

    for (int k = 0; k < D_H; k += 4) {
        const int ka = k + khalf;
        v2f a;
        a.x = (1.0f - ALPHA_C) * ar[ka]     + ALPHA_C * xr[ka];
        a.y = (1.0f - ALPHA_C) * ar[ka + 1] + ALPHA_C * xr[ka + 1];
#pragma unroll
        for (int t = 0; t < 4; ++t) {
            const int n = t * 16 + mlo;
            v2f b;
            b.x = W[(size_t)ka * D_H + n];
            b.y = W[(size_t)(ka + 1) * D_H + n];
            acc[t] = __builtin_amdgcn_wmma_f32_16x16x4_f32(
                false, a, false, b, (short)0, acc[t], false, false);
        }
    }

    const int mbase = row0 + (lane >> 4) * 8;
#pragma unroll
    for (int t = 0; t < 4; ++t) {
        const int n = t * 16 + mlo;
#pragma unroll
        for (int r = 0; r < 8; ++r) {
            int m = mbase + r;
            float v = acc[t][r];
            if (relu) v = fmaxf(v, 0.0f);
            if (m < N_NODES) hout[(size_t)m * D_H + n] = v;
        }
    }
}

// ---------------- aggregation --------------------------------------------

// acc[v][:] = dinv[v]^2 * h[v][:]   (self-loop term; also zero-fills acc)
__global__ void self_init_kernel(float* __restrict__ acc,
                                 const float* __restrict__ dinv,
                                 const float* __restrict__ h) {
    int i = blockIdx.x * blockDim.x + threadIdx.x;
    if (i < N_NODES * D_H) {
        int v = i >> 6;
        float d = dinv[v];
        acc[i] = d * d * h[i];
    }
}

// One wave per edge: lanes cover 64 features as two coalesced 128B chunks.
__global__ void __launch_bounds__(256)
edge_agg_kernel(float* __restrict__ acc, const float* __restrict__ dinv,
                const float* __restrict__ h, const int* __restrict__ row,
                const int* __restrict__ col, int e) {
    const int lane   = threadIdx.x & 31;
    const int gw     = (blockIdx.x * blockDim.x + threadIdx.x) >> 5;
    const int stride = (gridDim.x * blockDim.x) >> 5;
    for (int i = gw; i < e; i += stride) {
        const int r = row[i];
        const int c = col[i];
        const float nrm = dinv[r] * dinv[c];
        const float v0 = h[(size_t)r * D_H + lane];
        const float v1 = h[(size_t)r * D_H + 32 + lane];
        atomicAdd(acc + (size_t)c * D_H + lane,      nrm * v0);
        atomicAdd(acc + (size_t)c * D_H + 32 + lane, nrm * v1);
    }
}

// ---------------- output layer + log_softmax (wave per node) --------------

__global__ void __launch_bounds__(256)
out_kernel(const float* __restrict__ h, const float* __restrict__ Wout,
           const float* __restrict__ bout, float* __restrict__ out) {
    const int lane   = threadIdx.x & 31;
    const int gw     = (blockIdx.x * blockDim.x + threadIdx.x) >> 5;
    const int stride = (gridDim.x * blockDim.x) >> 5;
    for (int node = gw; node < N_NODES; node += stride) {
        const int c0 = lane;            // always < 47
        const int c1 = lane + 32;
        const bool v1ok = (c1 < D_OUT); // lanes 0..14
        float a0 = bout[c0];
        float a1 = v1ok ? bout[c1] : 0.0f;
        const float* hr = h + (size_t)node * D_H;
        const float* w0 = Wout + (size_t)c0 * D_H;
        const float* w1 = Wout + (size_t)(v1ok ? c1 : 0) * D_H;
        for (int k = 0; k < D_H; ++k) {
            const float hv = hr[k];
            a0 += hv * w0[k];
            a1 += hv * w1[k];
        }
        float mx = fmaxf(a0, v1ok ? a1 : -__builtin_inff());
#pragma unroll
        for (int off = 16; off > 0; off >>= 1)
            mx = fmaxf(mx, __shfl_xor(mx, off, 32));
        float s = expf(a0 - mx) + (v1ok ? expf(a1 - mx) : 0.0f);
#pragma unroll
        for (int off = 16; off > 0; off >>= 1)
            s += __shfl_xor(s, off, 32);
        const float lse = mx + logf(s);
        out[(size_t)node * D_OUT + c0] = a0 - lse;
        if (v1ok) out[(size_t)node * D_OUT + c1] = a1 - lse;
    }
}

// ---------------- driver --------------------------------------------------

extern "C" void kernel_launch(void* const* d_in, const int* in_sizes, int n_in,
                              void* d_out, int out_size, void* d_ws, size_t ws_size,
                              hipStream_t stream) {
    const float* x     = (const float*)d_in[0];
    const int*   edge  = (const int*)d_in[1];
    const float* Win   = (const float*)d_in[2];
    const float* bin   = (const float*)d_in[3];
    const float* Wcv   = (const float*)d_in[4];
    const float* Wout  = (const float*)d_in[5];
    const float* bout  = (const float*)d_in[6];
    float*       out   = (float*)d_out;

    const int E = in_sizes[1] / 2;
    const int* rowp = edge;        // edge_index[0] = source
    const int* colp = edge + E;    // edge_index[1] = target

    float* ws   = (float*)d_ws;
    float* dinv = ws;                                   // N floats (padded)
    float* x0   = ws + 100096;                          // N x 64
    float* accb = x0 + (size_t)N_NODES * D_H;           // N x 64
    float* hbuf = accb + (size_t)N_NODES * D_H;         // N x 64

    const dim3 blk(256);
    const int nb_nodes = (N_NODES + 255) / 256;
    const int nb_feat  = (N_NODES * D_H + 255) / 256;
    const int nb_gemm  = (N_NODES + 127) / 128;         // 8 waves x 16 rows
    const int nb_edge  = (E + 7) / 8;                   // 1 edge per wave
    const int nb_wavegrid = (N_NODES + 7) / 8;

    deg_init_kernel<<<nb_nodes, blk, 0, stream>>>(dinv);
    deg_count_kernel<<<(E + 255) / 256, blk, 0, stream>>>(dinv, colp, E);
    dinv_kernel<<<nb_nodes, blk, 0, stream>>>(dinv);

    gemm_in_kernel<<<nb_gemm, blk, 0, stream>>>(x, Win, bin, x0);

    for (int i = 0; i < N_LAYER; ++i) {
        const float* hsrc = (i == 0) ? x0 : hbuf;
        self_init_kernel<<<nb_feat, blk, 0, stream>>>(accb, dinv, hsrc);
        edge_agg_kernel<<<nb_edge, blk, 0, stream>>>(accb, dinv, hsrc, rowp, colp, E);
        gemm_layer_kernel<<<nb_gemm, blk, 0, stream>>>(
            accb, x0, Wcv + (size_t)i * D_H * D_H, hbuf, (i < N_LAYER - 1) ? 1 : 0);
    }

    out_kernel<<<nb_wavegrid, blk, 0, stream>>>(hbuf, Wout, bout, out);
}